// MyLinearLayer_68513318305821
// MI455X (gfx1250) — compile-verified
//
#include <hip/hip_runtime.h>
#include <hip/hip_bf16.h>

// ---------------------------------------------------------------------------
// out[n, h*7+k, c] = sum_w x[n,c,h,w] * weight[w*7+k, c] + bias[h*7+k, c]
// N=4096, C=640, H=W=7.  Memory-bound (AI ~1.75 FLOP/B, floor ~44us @23.3TB/s).
// Math on V_WMMA_F32_16X16X4_F32 (M=16 batch, K=7 split 4+3, N=7 of 16 cols).
// Tile: 16 n x 32 c per workgroup (16 wave32, each wave owns 2 channels).
//   Phase 1: GLOBAL_LOAD_ASYNC_TO_LDS_B128 stream of x (wave wv <-> row n=wv,
//            wave-uniform saddr); weight staged zero-padded in B-frag layout;
//            bias staged [49][32].  s_wait_asynccnt fence before barrier.
//   Phase 2: pure ds_load + 14 dense v_wmma per channel (C=0), then one
//            predicated block of immediate-offset ds_stores.
//   Phase 3: bias add + B128 coalesced stores, lanes along c.
// ---------------------------------------------------------------------------

typedef __attribute__((ext_vector_type(2))) float v2f;
typedef __attribute__((ext_vector_type(4))) float v4f;
typedef __attribute__((ext_vector_type(8))) float v8f;

#define TN 16
#define TC 32
#define NTHR 512
#define CSTR (640 * 49)       // x: floats between n images
#define XPITCH 1572           // x LDS pitch per n (1568+4: 16B-aligned, bank-spread)
#define OPITCH 36             // out-stage pitch per (n,hk) row (32+4)
#define WPITCH 33             // weight-frag pitch
#define XWORDS (TN * XPITCH)          // 25152
#define OWORDS (TN * 49 * OPITCH)     // 28224
#define WWORDS (8 * 16 * WPITCH)      // 4224  [wrow 0..7][k 0..15][cc]
#define BWORDS (49 * 32)              // 1568

__global__ __launch_bounds__(NTHR, 1)
void MyLinearLayer_wmma_kernel(const float* __restrict__ x,
                               const float* __restrict__ w,
                               const float* __restrict__ b,
                               float* __restrict__ out) {
    extern __shared__ float smem[];
    float* lds_x = smem;                             // 16B-aligned
    float* lds_o = smem + XWORDS;                    // 16B-aligned (25152*4)
    float* lds_w = smem + XWORDS + OWORDS;
    float* lds_b = smem + XWORDS + OWORDS + WWORDS;

    const int t    = threadIdx.x;
    const int lane = t & 31;
    const int wv   = t >> 5;              // wave id == batch row in tile
    const int n0   = blockIdx.x * TN;
    const int c0   = blockIdx.y * TC;

    // -------- Phase 1a: async B128 stream, wave wv owns row n = wv --------
    // Row = 32 channels' images, one contiguous 6272-B (392 float4) span.
    {
        const float* gsrc = x + (size_t)(n0 + wv) * CSTR + (size_t)c0 * 49;
        float* ldst = lds_x + wv * XPITCH;
#pragma unroll
        for (int j = 0; j < 12; ++j) {
            int q = lane + 32 * j;                       // float4 index
            unsigned la = (unsigned)(uintptr_t)(ldst + q * 4);
            int voff = q * 16;
            asm volatile("global_load_async_to_lds_b128 %0, %1, %2"
                         :: "v"(la), "v"(voff), "s"(gsrc) : "memory");
        }
        if (lane < 8) {                                  // 392 = 12*32 + 8
            int q = lane + 384;
            unsigned la = (unsigned)(uintptr_t)(ldst + q * 4);
            int voff = q * 16;
            asm volatile("global_load_async_to_lds_b128 %0, %1, %2"
                         :: "v"(la), "v"(voff), "s"(gsrc) : "memory");
        }
    }

    // -------- Phase 1b: weight tile, zero-padded B-fragment layout --------
    // lds_w[(wrow*16 + k)*WPITCH + cc]; wrow==7 / k>=7 entries are 0 (pad).
#pragma unroll
    for (int i = 0; i < 8; ++i) {
        int idx  = i * NTHR + t;                 // 0..4095
        int cc   = idx & 31;
        int k    = (idx >> 5) & 15;
        int wrow = idx >> 9;                     // 0..7
        int kc   = k < 7 ? k : 6;                // clamp -> in-bounds load
        int wc   = wrow < 7 ? wrow : 6;
        float v  = w[(wc * 7 + kc) * 640 + c0 + cc];
        v = (wrow < 7 && k < 7) ? v : 0.f;       // select, no exec branch
        lds_w[(wrow * 16 + k) * WPITCH + cc] = v;
    }

    // -------- Phase 1c: bias tile [hk][cc] --------------------------------
#pragma unroll
    for (int i = 0; i < 4; ++i) {
        int idx = i * NTHR + t;
        if (idx < BWORDS) {
            lds_b[idx] = b[(idx >> 5) * 640 + c0 + (idx & 31)];
        }
    }

    asm volatile("s_wait_asynccnt 0x0" ::: "memory");
    __syncthreads();

    // -------- Phase 2: WMMA (2 channels per wave, C=0, LDS-only) ----------
    // A 16x4 f32: lane L holds M=L%16; VGPR0 = K(2*hi), VGPR1 = K(2*hi+1)
    // B 4x16 f32: lane L holds N=L%16; VGPR0 = K(2*hi), VGPR1 = K(2*hi+1)
    // C/D 16x16 f32: VGPR j holds M = j + 8*hi, N = L%16
    const int n16 = lane & 15;
    const int hi  = lane >> 4;

#pragma unroll
    for (int ci = 0; ci < 2; ++ci) {
        const int cc = wv * 2 + ci;              // wave-uniform channel
        const float* wf = lds_w + (2 * hi * 16 + n16) * WPITCH + cc;
        v2f bf0, bf1;
        bf0.x = wf[0];                           // w = 2hi
        bf0.y = wf[16 * WPITCH];                 // w = 2hi+1
        bf1.x = wf[4 * 16 * WPITCH];             // w = 4+2hi
        bf1.y = wf[5 * 16 * WPITCH];             // w = 5+2hi (row 7 == 0 pad)

        const float* ax0 = lds_x + n16 * XPITCH + cc * 49;
        const float* axh = ax0 + 2 * hi;

        v8f accs[7];
#pragma unroll
        for (int h = 0; h < 7; ++h) {
            v2f a0, a1;
            a0.x = axh[h * 7 + 0];
            a0.y = axh[h * 7 + 1];
            a1.x = axh[h * 7 + 4];
            float v5 = ax0[h * 7 + 5];           // always in-bounds
            a1.y = hi ? 0.f : v5;                // cndmask, no exec branch

            v8f acc = {0.f, 0.f, 0.f, 0.f, 0.f, 0.f, 0.f, 0.f};
            acc = __builtin_amdgcn_wmma_f32_16x16x4_f32(
                false, a0, false, bf0, (short)0, acc, false, false);
            acc = __builtin_amdgcn_wmma_f32_16x16x4_f32(
                false, a1, false, bf1, (short)0, acc, false, false);
            accs[h] = acc;
        }

        // One predicated block; base reg + compile-time ds offsets (<64KB).
        if (n16 < 7) {
            float* po = lds_o + (hi * 8 * 49 + n16) * OPITCH + cc;
#pragma unroll
            for (int h = 0; h < 7; ++h) {
#pragma unroll
                for (int j = 0; j < 8; ++j) {
                    po[h * 7 * OPITCH + j * 49 * OPITCH] = accs[h][j];
                }
            }
        }
    }
    __syncthreads();

    // -------- Phase 3: bias add + B128 stores, wave wv owns row n = wv ----
    // 8 lanes x float4 = one contiguous 128-B row out[n, hk, c0:c0+32].
    {
        const float* lsrc = lds_o + wv * 49 * OPITCH;
        float* gout = out + ((size_t)(n0 + wv) * 49) * 640 + c0;
#pragma unroll
        for (int j = 0; j < 13; ++j) {
            int q = lane + 32 * j;               // 0..391 valid (49*8)
            if (j < 12 || lane < 8) {
                int hk = q >> 3;
                int cq = (q & 7) * 4;
                v4f v  = *(const v4f*)(lsrc + hk * OPITCH + cq);
                v4f bb = *(const v4f*)(lds_b + hk * 32 + cq);
                v = v + bb;
                *(v4f*)(gout + (size_t)hk * 640 + cq) = v;
            }
        }
    }
}

extern "C" void kernel_launch(void* const* d_in, const int* in_sizes, int n_in,
                              void* d_out, int out_size, void* d_ws, size_t ws_size,
                              hipStream_t stream) {
    const float* x = (const float*)d_in[0];   // [4096, 640, 7, 7]
    const float* w = (const float*)d_in[1];   // [49, 640]
    const float* b = (const float*)d_in[2];   // [49, 640]
    float* out = (float*)d_out;               // [4096, 49, 640]

    const size_t shmem =
        (size_t)(XWORDS + OWORDS + WWORDS + BWORDS) * sizeof(float); // ~231 KB
    (void)hipFuncSetAttribute((const void*)MyLinearLayer_wmma_kernel,
                              hipFuncAttributeMaxDynamicSharedMemorySize,
                              (int)shmem);

    dim3 grid(4096 / TN, 640 / TC);           // 256 x 20, exact cover
    MyLinearLayer_wmma_kernel<<<grid, NTHR, shmem, stream>>>(x, w, b, out);
}